// SelfAttention_3770981286169
// MI455X (gfx1250) — compile-verified
//
#include <hip/hip_runtime.h>

typedef __attribute__((ext_vector_type(2))) float v2f;
typedef __attribute__((ext_vector_type(8))) float v8f;

#define S_LEN 2048
#define DHEAD 64
#define QTILE 128          // queries per block (8 waves x 16)
#define WQ    16           // queries per wave
#define KTILE 16           // keys per LDS tile
#define KSTR  68           // LDS row stride for roped-K tile (conflict-free frag reads)
#define VSTR  68           // LDS row stride for V^T tile (conflict-free frag reads)

__device__ __forceinline__ v8f wmma_f32(v2f a, v2f b, v8f c) {
  // D = A(16x4 f32) * B(4x16 f32) + C(16x16 f32)
  return __builtin_amdgcn_wmma_f32_16x16x4_f32(
      false, a, false, b, (short)0, c, false, false);
}

// RoPE an A/B fragment in place: lane holds row `pos`, cols 4c+2*half (+1);
// pair (j, j+32) is lane-local. Folds `scale`.
__device__ __forceinline__ void rope16(v2f* f, float pos, int half, float scale) {
  const float NL = -0.2878231366f;  // -ln(10000)/32
#pragma unroll
  for (int c = 0; c < 8; ++c) {
    const int j = 4 * c + 2 * half;
    float a0 = pos * __expf(NL * (float)j);
    float a1 = pos * __expf(NL * (float)(j + 1));
    float s0, c0, s1, c1;
    __sincosf(a0, &s0, &c0);
    __sincosf(a1, &s1, &c1);
    float x1a = f[c].x, x2a = f[c + 8].x;
    float x1b = f[c].y, x2b = f[c + 8].y;
    f[c].x     = (x1a * c0 - x2a * s0) * scale;
    f[c + 8].x = (x1a * s0 + x2a * c0) * scale;
    f[c].y     = (x1b * c1 - x2b * s1) * scale;
    f[c + 8].y = (x1b * s1 + x2b * c1) * scale;
  }
}

// Loader-side RoPE for one K row-pair-column: rotate (x1, x2) at columns
// (pb, pb+1) / (pb+32, pb+33) and store into the LDS K tile row.
__device__ __forceinline__ void rope_store_k(float* dstrow, v2f x1, v2f x2,
                                             float pos, float inv0, float inv1,
                                             int pb) {
  float s0, c0, s1, c1;
  __sincosf(pos * inv0, &s0, &c0);
  __sincosf(pos * inv1, &s1, &c1);
  v2f lo, hi;
  lo.x = x1.x * c0 - x2.x * s0;  hi.x = x1.x * s0 + x2.x * c0;
  lo.y = x1.y * c1 - x2.y * s1;  hi.y = x1.y * s1 + x2.y * c1;
  *(v2f*)(dstrow + pb)      = lo;
  *(v2f*)(dstrow + pb + 32) = hi;
}

__global__ __launch_bounds__(256) void rope_flash_attn_f32(
    const float* __restrict__ Q, const float* __restrict__ K,
    const float* __restrict__ V, float* __restrict__ O) {
  __shared__ float kbuf[2][KTILE * KSTR];    // roped K tile, row = key
  __shared__ float vtbuf[2][DHEAD * VSTR];   // V^T tile,    row = d, col = key

  const int qt   = blockIdx.x;   // 0..15
  const int bh   = blockIdx.y;   // 0..63
  const int tid  = threadIdx.x;
  const int wave = tid >> 5;
  const int lane = tid & 31;
  const int lrow = lane & 15;
  const int half = lane >> 4;

  const int qw = qt * QTILE + wave * WQ;
  const size_t head_off = (size_t)bh * S_LEN * DHEAD;

  // ---- Q fragment (acts as WMMA B operand for S^T = K * Q^T), RoPE, 1/8 ----
  v2f qv[16];
  {
    const float* qrow = Q + head_off + (size_t)(qw + lrow) * DHEAD + 2 * half;
#pragma unroll
    for (int c = 0; c < 16; ++c) qv[c] = *(const v2f*)(qrow + 4 * c);
    rope16(qv, (float)(qw + lrow), half, 0.125f);
  }

  const v8f zero8 = {0.f, 0.f, 0.f, 0.f, 0.f, 0.f, 0.f, 0.f};
  v8f acc[4] = {zero8, zero8, zero8, zero8};   // O^T: acc[n][r] = O[q][16n+8*half+r]
  float m_run = -__builtin_inff();             // per-lane: this lane's query stats
  float l_run = 0.f;

  // ---- cooperative loader indices (all 256 threads) ----
  const int krow = tid >> 4;            // K tile row (key)     0..15
  const int kpb  = (tid & 15) * 2;      // K pair column j0     0,2,..,30
  const int vd   = tid & 63;            // V^T row (d)          0..63
  const int vkg  = tid >> 6;            // V key group          0..3
  const float NL = -0.2878231366f;
  const float kinv0 = __expf(NL * (float)kpb);
  const float kinv1 = __expf(NL * (float)(kpb + 1));

  const int nkt = 8 * (qt + 1);

  // ---- prologue: stage tile 0 ----
  {
    const float* kr = K + head_off + (size_t)krow * DHEAD;
    v2f x1 = *(const v2f*)(kr + kpb);
    v2f x2 = *(const v2f*)(kr + kpb + 32);
    const float* vb = V + head_off + (size_t)(4 * vkg) * DHEAD + vd;
    float4 vt;
    vt.x = vb[0 * DHEAD]; vt.y = vb[1 * DHEAD];
    vt.z = vb[2 * DHEAD]; vt.w = vb[3 * DHEAD];
    rope_store_k(&kbuf[0][krow * KSTR], x1, x2, (float)krow, kinv0, kinv1, kpb);
    *(float4*)&vtbuf[0][vd * VSTR + 4 * vkg] = vt;
  }
  __syncthreads();

  for (int kt = 0; kt < nkt; ++kt) {
    const int cur = kt & 1, nxt = cur ^ 1;
    const int ktb = kt * KTILE;

    // ---- prefetch next tile into registers (overlaps compute) ----
    const bool more = (kt + 1) < nkt;
    v2f x1, x2; float4 vt;
    if (more) {
      const float* kr = K + head_off + (size_t)(ktb + KTILE + krow) * DHEAD;
      x1 = *(const v2f*)(kr + kpb);
      x2 = *(const v2f*)(kr + kpb + 32);
      const float* vb = V + head_off + (size_t)(ktb + KTILE + 4 * vkg) * DHEAD + vd;
      vt.x = vb[0 * DHEAD]; vt.y = vb[1 * DHEAD];
      vt.z = vb[2 * DHEAD]; vt.w = vb[3 * DHEAD];
    }

    // ---- compute (wave-uniform causal skip) ----
    if (ktb < qw + WQ) {
      // K A-fragments (already RoPE'd by loader)
      v2f kv[16];
      const float* krow_s = &kbuf[cur][lrow * KSTR + 2 * half];
#pragma unroll
      for (int c = 0; c < 16; ++c) kv[c] = *(const v2f*)(krow_s + 4 * c);

      // S^T = K_r * Q_r^T : C layout -> lane owns query q = qw+lrow,
      // keys ktb + r + 8*half live in the 8 accumulator slots.
      v8f s = zero8;
#pragma unroll
      for (int c = 0; c < 16; ++c) s = wmma_f32(kv[c], qv[c], s);

      const int q_g = qw + lrow;
      float sv[8];
      if (ktb == qw) {  // only the diagonal tile needs masking
#pragma unroll
        for (int r = 0; r < 8; ++r)
          sv[r] = ((ktb + r + 8 * half) <= q_g) ? s[r] : -__builtin_inff();
      } else {
#pragma unroll
        for (int r = 0; r < 8; ++r) sv[r] = s[r];
      }

      // per-lane online softmax (one query per lane; halves combined once)
      float mx = sv[0];
#pragma unroll
      for (int r = 1; r < 8; ++r) mx = fmaxf(mx, sv[r]);
      mx = fmaxf(mx, __shfl_xor(mx, 16, 32));
      const float m_new = fmaxf(m_run, mx);
      const float corr = __expf(m_run - m_new);
      m_run = m_new;
      float p[8], rs = 0.f;
#pragma unroll
      for (int r = 0; r < 8; ++r) { p[r] = __expf(sv[r] - m_new); rs += p[r]; }
      rs += __shfl_xor(rs, 16, 32);
      l_run = l_run * corr + rs;

      // rescale running O^T accumulator (per-lane scalar!)
#pragma unroll
      for (int n = 0; n < 4; ++n)
#pragma unroll
        for (int r = 0; r < 8; ++r) acc[n][r] *= corr;

      // P^T C-layout -> B-layout via cross-half exchange (no LDS bounce)
      float po[8];
#pragma unroll
      for (int r = 0; r < 8; ++r) po[r] = __shfl_xor(p[r], 16, 32);
      v2f pB[4];
      pB[0].x = half ? po[2] : p[0];  pB[0].y = half ? po[3] : p[1];
      pB[1].x = half ? po[6] : p[4];  pB[1].y = half ? po[7] : p[5];
      pB[2].x = half ? p[2] : po[0];  pB[2].y = half ? p[3] : po[1];
      pB[3].x = half ? p[6] : po[4];  pB[3].y = half ? p[7] : po[5];

      // O^T += V^T * P^T : A = V^T tile-n K-chunk (aligned v2f LDS loads)
      const float* vbs = vtbuf[cur];
#pragma unroll
      for (int n = 0; n < 4; ++n) {
        const float* vr = vbs + (size_t)(16 * n + lrow) * VSTR + 2 * half;
#pragma unroll
        for (int c = 0; c < 4; ++c) {
          v2f a = *(const v2f*)(vr + 4 * c);
          acc[n] = wmma_f32(a, pB[c], acc[n]);
        }
      }
    }

    // ---- stage prefetched tile (RoPE K here, once per block) ----
    if (more) {
      rope_store_k(&kbuf[nxt][krow * KSTR], x1, x2,
                   (float)(ktb + KTILE + krow), kinv0, kinv1, kpb);
      *(float4*)&vtbuf[nxt][vd * VSTR + 4 * vkg] = vt;
    }
    __syncthreads();
  }

  // ---- epilogue: normalize (per-lane scalar) and store O ----
  const float invl = 1.0f / l_run;
  float* orow = O + head_off + (size_t)(qw + lrow) * DHEAD + 8 * half;
#pragma unroll
  for (int n = 0; n < 4; ++n) {
    float4 lo, hi;
    lo.x = acc[n][0] * invl; lo.y = acc[n][1] * invl;
    lo.z = acc[n][2] * invl; lo.w = acc[n][3] * invl;
    hi.x = acc[n][4] * invl; hi.y = acc[n][5] * invl;
    hi.z = acc[n][6] * invl; hi.w = acc[n][7] * invl;
    *(float4*)(orow + 16 * n)     = lo;
    *(float4*)(orow + 16 * n + 4) = hi;
  }
}

extern "C" void kernel_launch(void* const* d_in, const int* in_sizes, int n_in,
                              void* d_out, int out_size, void* d_ws, size_t ws_size,
                              hipStream_t stream) {
  (void)in_sizes; (void)n_in; (void)d_ws; (void)ws_size; (void)out_size;
  const float* Q = (const float*)d_in[0];
  const float* K = (const float*)d_in[1];
  const float* V = (const float*)d_in[2];
  // d_in[3] (mask) ignored: causal structure computed analytically.
  float* O = (float*)d_out;
  dim3 grid(S_LEN / QTILE, 4 * 16);  // 16 query tiles x 64 (b,h)
  rope_flash_attn_f32<<<grid, 256, 0, stream>>>(Q, K, V, O);
}